// GCNConv_46299747451336
// MI455X (gfx1250) — compile-verified
//
#include <hip/hip_runtime.h>

typedef __attribute__((ext_vector_type(2))) float v2f;
typedef __attribute__((ext_vector_type(8))) float v8f;

#define DFEAT 64
#define LDSW_STRIDE 66   // pad 64-float rows: keeps 8B alignment, breaks 256B bank period

// ---------------- degree / norm ----------------

__global__ void k_zero_f32(float* p, int n) {
    int i = blockIdx.x * blockDim.x + threadIdx.x;
    if (i < n) p[i] = 0.0f;
}

__global__ void k_degree(const int* __restrict__ row, float* __restrict__ deg, int E) {
    int e = blockIdx.x * blockDim.x + threadIdx.x;
    if (e < E) atomicAdd(&deg[row[e]], 1.0f);
}

__global__ void k_dinv(float* __restrict__ deg, int n) {
    int i = blockIdx.x * blockDim.x + threadIdx.x;
    if (i < n) deg[i] = rsqrtf(deg[i] + 1.0f);   // +1 = self loop
}

// ---------------- h = x @ W^T + b via V_WMMA_F32_16X16X4_F32 ----------------
// Block: 128 threads = 4 waves; each wave computes a 16-row strip of h,
// covering all 64 output columns as four 16x16 accumulator tiles.
__global__ void __launch_bounds__(128)
k_gemm_wmma(const float* __restrict__ x, const float* __restrict__ W,
            const float* __restrict__ bias, float* __restrict__ h, int n) {
    __shared__ float Ws[DFEAT * LDSW_STRIDE];

    const int tid = threadIdx.x;
    // stage W (64x64 f32 = 16 KB) into LDS, padded
    for (int i = tid; i < DFEAT * DFEAT; i += 128) {
        int r = i >> 6, c = i & 63;
        Ws[r * LDSW_STRIDE + c] = W[i];
    }
    __syncthreads();

    const int wave = tid >> 5;          // 0..3
    const int lane = tid & 31;
    const int lg   = lane >> 4;         // half-wave group (K sub-pair select)
    const int ll   = lane & 15;         // M (for A) / N (for B) within tile
    const int m0   = blockIdx.x * 64 + wave * 16;

    v8f acc0 = {}, acc1 = {}, acc2 = {}, acc3 = {};

    const int m      = m0 + ll;
    const int msafe  = (m < n) ? m : (n - 1);
    const bool mok   = (m < n);
    const float* xr  = x + (long long)msafe * DFEAT;

#pragma unroll
    for (int k0 = 0; k0 < DFEAT; k0 += 4) {
        const int kk = k0 + lg * 2;
        // A fragment: 16x4 f32, lane ll = row M, VGPR pair = K {kk, kk+1}
        v2f a = *(const v2f*)(xr + kk);
        if (!mok) { a.x = 0.0f; a.y = 0.0f; }   // branchless lane select, EXEC stays full
        // B fragments: 4x16 f32, lane ll = column N, from LDS
        v2f b0 = *(const v2f*)(Ws + (0 * 16 + ll) * LDSW_STRIDE + kk);
        v2f b1 = *(const v2f*)(Ws + (1 * 16 + ll) * LDSW_STRIDE + kk);
        v2f b2 = *(const v2f*)(Ws + (2 * 16 + ll) * LDSW_STRIDE + kk);
        v2f b3 = *(const v2f*)(Ws + (3 * 16 + ll) * LDSW_STRIDE + kk);

        acc0 = __builtin_amdgcn_wmma_f32_16x16x4_f32(false, a, false, b0, (short)0, acc0, false, false);
        acc1 = __builtin_amdgcn_wmma_f32_16x16x4_f32(false, a, false, b1, (short)0, acc1, false, false);
        acc2 = __builtin_amdgcn_wmma_f32_16x16x4_f32(false, a, false, b2, (short)0, acc2, false, false);
        acc3 = __builtin_amdgcn_wmma_f32_16x16x4_f32(false, a, false, b3, (short)0, acc3, false, false);
    }

    const float bi0 = bias[ 0 + ll];
    const float bi1 = bias[16 + ll];
    const float bi2 = bias[32 + ll];
    const float bi3 = bias[48 + ll];

    // C/D layout: VGPR j -> M = j + 8*lg, N = ll (+ 16*tile)
#pragma unroll
    for (int j = 0; j < 8; ++j) {
        int ms = m0 + j + 8 * lg;
        if (ms < n) {
            float* hr = h + (long long)ms * DFEAT;
            hr[ 0 + ll] = acc0[j] + bi0;
            hr[16 + ll] = acc1[j] + bi1;
            hr[32 + ll] = acc2[j] + bi2;
            hr[48 + ll] = acc3[j] + bi3;
        }
    }
}

// ---------------- out init with self-loop term: out[i] = dinv[i]^2 * h[i] ----------------
__global__ void k_self_init(const float* __restrict__ h, const float* __restrict__ dinv,
                            float* __restrict__ out, int n) {
    int idx = blockIdx.x * blockDim.x + threadIdx.x;
    if (idx < n * DFEAT) {
        int i = idx >> 6;
        float di = dinv[i];
        out[idx] = di * di * h[idx];
    }
}

// ---------------- edge scatter: out[row] += dinv[row]*dinv[col] * h[col] ----------------
__global__ void k_scatter(const int* __restrict__ row, const int* __restrict__ col,
                          const float* __restrict__ h, const float* __restrict__ dinv,
                          float* __restrict__ out, int E) {
    int idx = blockIdx.x * blockDim.x + threadIdx.x;   // E*64 = 51.2M < 2^31
    if (idx >= E * DFEAT) return;
    int e = idx >> 6;
    int f = idx & 63;
    int r = row[e];
    int c = col[e];
    float norm = dinv[r] * dinv[c];
    atomicAdd(&out[r * DFEAT + f], norm * h[c * DFEAT + f]);
}

// ---------------- launch ----------------

extern "C" void kernel_launch(void* const* d_in, const int* in_sizes, int n_in,
                              void* d_out, int out_size, void* d_ws, size_t ws_size,
                              hipStream_t stream) {
    const float* x  = (const float*)d_in[0];
    const int*   ei = (const int*)d_in[1];
    const float* W  = (const float*)d_in[2];
    const float* b  = (const float*)d_in[3];
    float*       out = (float*)d_out;

    const int N = in_sizes[0] / DFEAT;   // 50000
    const int E = in_sizes[1] / 2;       // 800000
    const int* row = ei;
    const int* col = ei + E;

    // workspace layout: deg/dinv [N floats, rounded up], then h [N*64 floats]
    float* deg = (float*)d_ws;
    float* h   = deg + ((N + 255) & ~255);

    k_zero_f32<<<(N + 255) / 256, 256, 0, stream>>>(deg, N);
    k_degree  <<<(E + 255) / 256, 256, 0, stream>>>(row, deg, E);
    k_dinv    <<<(N + 255) / 256, 256, 0, stream>>>(deg, N);

    int gemm_blocks = (N + 63) / 64;
    k_gemm_wmma<<<gemm_blocks, 128, 0, stream>>>(x, W, b, h, N);

    k_self_init<<<(N * DFEAT + 255) / 256, 256, 0, stream>>>(h, deg, out, N);

    long long work = (long long)E * DFEAT;
    int sc_blocks = (int)((work + 255) / 256);
    k_scatter<<<sc_blocks, 256, 0, stream>>>(row, col, h, deg, out, E);
}